// SampleLoss_70480413328151
// MI455X (gfx1250) — compile-verified
//
#include <hip/hip_runtime.h>
#include <math.h>

typedef float v2f __attribute__((ext_vector_type(2)));
typedef float v8f __attribute__((ext_vector_type(8)));

#define NBATCH   2
#define ROWTILES 4                 // 64 sampled rows per block
#define FINF_BITS 0x7F800000u

// ---------------------------------------------------------------------------
// Kernel 1: init workspace min-arrays to +inf (bit pattern 0x7F800000)
// ---------------------------------------------------------------------------
__global__ void init_mins_kernel(unsigned int* __restrict__ ws, size_t n) {
    size_t i = (size_t)blockIdx.x * blockDim.x + threadIdx.x;
    if (i < n) ws[i] = FINF_BITS;
}

// ---------------------------------------------------------------------------
// Kernel 2: WMMA squared-distance tiles.
// One wave per block; ROWTILES*16 sampled rows (blockIdx.y) x
// colTilesPerBlock column tiles of 16 raw points (blockIdx.x); b=blockIdx.z.
//
// K=4 fragment carries the B-norm and the -2 factor:
//   A row m  = ( ax,  ay,  az,  1    )
//   B col n  = (-2bx, -2by, -2bz, |b|^2)
//   D[m][n]  = |b_n|^2 - 2 a_m.b_n  ;  sq = |a_m|^2 + D[m][n]
// Minima tracked on SQUARED distances; sqrt deferred to finalize
// (monotonic => commutes with min/max).
//
// Per column tile: ONE B load + fragment build feeds ROWTILES WMMAs; the
// column atomic-min covers all ROWTILES*16 rows and is issued by ALL lanes
// (both halves hit the same address; atomic min is commutative) so the loop
// body has no EXEC manipulation and no DS ops.
// ---------------------------------------------------------------------------
__global__ __launch_bounds__(32) void chamfer_tile_kernel(
    const float* __restrict__ sampled,           // [B][N][4]
    const float* __restrict__ raw,               // [B][M][4]
    unsigned int* __restrict__ minOverSampled,   // [B][M] (f32 bits of d^2)
    unsigned int* __restrict__ minOverRaw,       // [B][N] (f32 bits of d^2)
    int N, int M, int colTilesPerBlock)
{
    const int lane = threadIdx.x;     // 0..31
    const int l16  = lane & 15;
    const int half = lane >> 4;       // 0 or 1
    const int b    = blockIdx.z;
    const int rowBase = blockIdx.y * (16 * ROWTILES);
    const int ctStart = blockIdx.x * colTilesPerBlock;

    const float* S = sampled + (size_t)b * N * 4;
    const float* R = raw     + (size_t)b * M * 4;

    // ---- A fragments + per-lane row norms (loop invariant) ----
    v2f   afrag[ROWTILES];
    float rn[ROWTILES][8];
#pragma unroll
    for (int rt = 0; rt < ROWTILES; ++rt) {
        const float* srow = S + (size_t)(rowBase + rt * 16 + l16) * 4;
        float s0 = srow[0], s1 = srow[1], s2 = srow[2];
        afrag[rt].x = half ? s2   : s0;   // K2 : K0
        afrag[rt].y = half ? 1.0f : s1;   // K3 : K1 (K3 = 1 multiplies |b|^2)
        float normA = s0 * s0 + s1 * s1 + s2 * s2;
#pragma unroll
        for (int v = 0; v < 8; ++v)
            rn[rt][v] = __shfl(normA, half * 8 + v, 32);
    }

    float rowMin[ROWTILES][8];
#pragma unroll
    for (int rt = 0; rt < ROWTILES; ++rt)
#pragma unroll
        for (int v = 0; v < 8; ++v) rowMin[rt][v] = __builtin_inff();

#pragma unroll 2
    for (int ct = ctStart; ct < ctStart + colTilesPerBlock; ++ct) {
        const int col = ct * 16 + l16;
        const float* rrow = R + (size_t)col * 4;
        float r0 = rrow[0], r1 = rrow[1], r2 = rrow[2];
        float normB = r0 * r0 + r1 * r1 + r2 * r2;

        v2f bfrag;
        bfrag.x = half ? (-2.0f * r2) : (-2.0f * r0);  // K2 : K0
        bfrag.y = half ? normB        : (-2.0f * r1);  // K3 : K1

        float colMin = __builtin_inff();
#pragma unroll
        for (int rt = 0; rt < ROWTILES; ++rt) {
            v8f c = {};
            c = __builtin_amdgcn_wmma_f32_16x16x4_f32(
                    false, afrag[rt], false, bfrag,
                    (short)0, c, false, false);
#pragma unroll
            for (int v = 0; v < 8; ++v) {
                float sq = fmaxf(rn[rt][v] + c[v], 0.0f);  // squared distance
                rowMin[rt][v] = fminf(rowMin[rt][v], sq);
                colMin        = fminf(colMin, sq);
            }
        }
        // all 32 lanes issue the atomic (halves alias the same address);
        // min is commutative so no cross-half shuffle / exec branch needed
        atomicMin(&minOverSampled[(size_t)b * M + col],
                  __float_as_uint(colMin));
    }

    // reduce rowMin across the 16 columns in each half, then write
#pragma unroll
    for (int rt = 0; rt < ROWTILES; ++rt) {
#pragma unroll
        for (int v = 0; v < 8; ++v) {
            float m = rowMin[rt][v];
            m = fminf(m, __shfl_xor(m, 8, 32));
            m = fminf(m, __shfl_xor(m, 4, 32));
            m = fminf(m, __shfl_xor(m, 2, 32));
            m = fminf(m, __shfl_xor(m, 1, 32));
            rowMin[rt][v] = m;
        }
    }
    if (l16 == 0) {
#pragma unroll
        for (int rt = 0; rt < ROWTILES; ++rt)
#pragma unroll
            for (int v = 0; v < 8; ++v) {
                int row = rowBase + rt * 16 + half * 8 + v;
                atomicMin(&minOverRaw[(size_t)b * N + row],
                          __float_as_uint(rowMin[rt][v]));
            }
    }
}

// ---------------------------------------------------------------------------
// Kernel 3: finalize. Arrays hold SQUARED min distances; sqrt applied here.
//   loss_b = 5*mean(sqrt(mos_b)) + mean(sqrt(mor_b)) + max(sqrt(mor_b))
//   out    = mean_b(loss_b)
// ---------------------------------------------------------------------------
__global__ __launch_bounds__(256) void finalize_kernel(
    const unsigned int* __restrict__ minOverSampled,
    const unsigned int* __restrict__ minOverRaw,
    float* __restrict__ out, int N, int M)
{
    __shared__ float red[256];
    const int tid = threadIdx.x;
    float result = 0.0f;

    for (int b = 0; b < NBATCH; ++b) {
        float s = 0.0f;
        for (int i = tid; i < M; i += 256) {
            float v = __uint_as_float(minOverSampled[(size_t)b * M + i]);
            s += sqrtf(fmaxf(v, 0.0f));
        }
        red[tid] = s; __syncthreads();
        for (int st = 128; st >= 1; st >>= 1) {
            if (tid < st) red[tid] += red[tid + st];
            __syncthreads();
        }
        float sumS = red[0]; __syncthreads();

        float r = 0.0f, mx = 0.0f;
        for (int i = tid; i < N; i += 256) {
            float v = __uint_as_float(minOverRaw[(size_t)b * N + i]);
            float d = sqrtf(fmaxf(v, 0.0f));
            r += d;
            mx = fmaxf(mx, d);
        }
        red[tid] = r; __syncthreads();
        for (int st = 128; st >= 1; st >>= 1) {
            if (tid < st) red[tid] += red[tid + st];
            __syncthreads();
        }
        float sumR = red[0]; __syncthreads();

        red[tid] = mx; __syncthreads();
        for (int st = 128; st >= 1; st >>= 1) {
            if (tid < st) red[tid] = fmaxf(red[tid], red[tid + st]);
            __syncthreads();
        }
        float maxR = red[0]; __syncthreads();

        result += 5.0f * (sumS / (float)M) + (sumR / (float)N) + maxR;
    }

    if (tid == 0) out[0] = result / (float)NBATCH;
}

// ---------------------------------------------------------------------------
extern "C" void kernel_launch(void* const* d_in, const int* in_sizes, int n_in,
                              void* d_out, int out_size, void* d_ws, size_t ws_size,
                              hipStream_t stream) {
    const float* sampled = (const float*)d_in[0]; // [B][N][4]
    const float* raw     = (const float*)d_in[1]; // [B][M][4]

    const int B = NBATCH;
    const int N = in_sizes[0] / (B * 4);   // 4096
    const int M = in_sizes[1] / (B * 4);   // 16384

    unsigned int* mos = (unsigned int*)d_ws;          // [B][M]
    unsigned int* mor = mos + (size_t)B * M;          // [B][N]
    size_t totalMins = (size_t)B * M + (size_t)B * N;

    init_mins_kernel<<<(unsigned)((totalMins + 255) / 256), 256, 0, stream>>>(
        mos, totalMins);

    const int colTilesPerBlock = 32;                   // 512 columns per block
    const int colBlocks = (M / 16) / colTilesPerBlock; // 32
    dim3 grid(colBlocks, N / (16 * ROWTILES), B);      // 32 x 64 x 2 wave32 blocks
    chamfer_tile_kernel<<<grid, 32, 0, stream>>>(
        sampled, raw, mos, mor, N, M, colTilesPerBlock);

    finalize_kernel<<<1, 256, 0, stream>>>(mos, mor, (float*)d_out, N, M);
}